// KMeans_15771119911643
// MI455X (gfx1250) — compile-verified
//
#include <hip/hip_runtime.h>
#include <hip/hip_bf16.h>

typedef float v2f __attribute__((ext_vector_type(2)));
typedef float v8f __attribute__((ext_vector_type(8)));
typedef int   v4i __attribute__((ext_vector_type(4)));

// Typed address-space pointers for the async global->LDS builtin:
// signature (from compiler diagnostic): (v4i AS1* src, v4i AS3* dst, Ii off, Ii cpol)
typedef __attribute__((address_space(1))) v4i* as1_v4i_p;
typedef __attribute__((address_space(3))) v4i* as3_v4i_p;

// Problem constants (from setup_inputs): B=16, N=32768, F=128, K=32, iter=10
constexpr int BB = 16;
constexpr int NN = 32768;
constexpr int FF = 128;
constexpr int KK = 32;
constexpr int ITERS = 10;

constexpr int TILE_ROWS  = 64;   // rows per staged LDS tile (4 waves x 16)
constexpr int LDS_STRIDE = 132;  // F + 4 pad: 16B-aligned rows, conflict-free frag reads
constexpr int TILES_PER_BLOCK = 8;   // 512 rows per block -> amortized atomic flush
constexpr int BLOCKS_X = NN / (TILE_ROWS * TILES_PER_BLOCK);  // 64

// --- CDNA5 async global->LDS staging (guarded: toolchains differ) -----------
#if defined(__has_builtin)
#  if __has_builtin(__builtin_amdgcn_global_load_async_to_lds_b128)
#    define HAVE_ASYNC_LDS 1
#  endif
#endif
#ifndef HAVE_ASYNC_LDS
#  define HAVE_ASYNC_LDS 0
#endif

__device__ __forceinline__ void wait_asynccnt_zero()
{
#if defined(__has_builtin) && __has_builtin(__builtin_amdgcn_s_wait_asynccnt)
    __builtin_amdgcn_s_wait_asynccnt(0);
#else
    asm volatile("s_wait_asynccnt 0" ::: "memory");
#endif
}

// Stage a 64x128 f32 tile (row-major, padded stride) from global into LDS.
__device__ __forceinline__ void stage_tile(const float* __restrict__ gsrc,
                                           float* __restrict__ lds, int tid)
{
#if HAVE_ASYNC_LDS
    // Direct global->LDS copies: no VGPR round trip, tracked by ASYNCcnt.
    float* gmut = const_cast<float*>(gsrc);
    #pragma unroll
    for (int j = 0; j < (TILE_ROWS * (FF / 4)) / 128; ++j) {   // 16 b128 ops/lane
        const int i = tid + j * 128;
        const int r = i >> 5, c4 = i & 31;                      // FF/4 == 32
        __builtin_amdgcn_global_load_async_to_lds_b128(
            (as1_v4i_p)(gmut + (size_t)i * 4),
            (as3_v4i_p)(lds + r * LDS_STRIDE + c4 * 4),
            0, 0);
    }
    wait_asynccnt_zero();   // this wave's portion done; barrier covers the rest
#else
    const float4* g4 = (const float4*)gsrc;
    #pragma unroll 4
    for (int i = tid; i < TILE_ROWS * (FF / 4); i += 128) {
        const int r = i >> 5, c4 = i & 31;
        float4 v = g4[r * (FF / 4) + c4];
        float* dst = &lds[r * LDS_STRIDE + c4 * 4];
        dst[0] = v.x; dst[1] = v.y; dst[2] = v.z; dst[3] = v.w;
    }
#endif
}

// ---------------------------------------------------------------------------
// Gather initial centroids and their squared norms.
__global__ __launch_bounds__(128)
void init_centroids_kernel(const float* __restrict__ data,
                           const int*   __restrict__ ids,
                           float* __restrict__ cent,
                           float* __restrict__ c2)
{
    const int b = blockIdx.y, k = blockIdx.x, t = threadIdx.x;
    const int id = ids[b * KK + k];
    const float v = data[((size_t)b * NN + id) * FF + t];
    cent[((size_t)b * KK + k) * FF + t] = v;

    float p = v * v;
    __shared__ float red[4];
    #pragma unroll
    for (int off = 16; off > 0; off >>= 1) p += __shfl_xor(p, off, 32);
    if ((t & 31) == 0) red[t >> 5] = p;
    __syncthreads();
    if (t == 0) c2[b * KK + k] = red[0] + red[1] + red[2] + red[3];
}

// ---------------------------------------------------------------------------
// Fused pass: cid = argmin_k (c2[k] - 2<x,c_k>)  (WMMA GEMM #1, x2/sqrt dropped:
// both monotone per-point), then sums += one_hot(cid)^T * data (WMMA GEMM #2,
// A-fragments synthesized from the just-computed cids). Data read once per pass.
template<bool ACCUM>
__global__ __launch_bounds__(128)
void fused_assign_accum_kernel(const float* __restrict__ data,   // [B][N][F]
                               const float* __restrict__ cent,   // [B][K][F]
                               const float* __restrict__ c2,     // [B][K]
                               int*   __restrict__ out,          // [B][N]
                               float* __restrict__ gsums,        // [B][K][F]
                               float* __restrict__ gcounts)      // [B][K]
{
    __shared__ float sData[TILE_ROWS * LDS_STRIDE];  // 33792 B
    __shared__ float sCent[KK * LDS_STRIDE];         // 16896 B
    __shared__ float sC2[KK];
    __shared__ int   sCid[TILE_ROWS];
    __shared__ float sCnt[KK];

    const int b    = blockIdx.y;
    const int tid  = threadIdx.x;
    const int lane = tid & 31;
    const int wave = tid >> 5;
    const int half = lane >> 4;      // 0: lanes 0-15 hold K=0,1 ; 1: lanes 16-31 hold K=2,3
    const int l15  = lane & 15;
    const int koff = half * 2;

    // Stage centroids + c2 once per block; zero count histogram.
    const float4* gcent = (const float4*)(cent + (size_t)b * KK * FF);
    #pragma unroll 4
    for (int i = tid; i < KK * (FF / 4); i += 128) {
        const int r = i >> 5, c4 = i & 31;
        float4 v = gcent[r * (FF / 4) + c4];
        float* dst = &sCent[r * LDS_STRIDE + c4 * 4];
        dst[0] = v.x; dst[1] = v.y; dst[2] = v.z; dst[3] = v.w;
    }
    if (tid < KK) { sC2[tid] = c2[b * KK + tid]; sCnt[tid] = 0.0f; }

    // Per-wave accumulation C-tiles: [cluster half][feature tile] of 16x16,
    // wave w owns features [32w, 32w+32). Live across all tiles of the block.
    v8f accS[2][2] = {};

    const int tileBase = blockIdx.x * (TILE_ROWS * TILES_PER_BLOCK);

    for (int t = 0; t < TILES_PER_BLOCK; ++t) {
        const int row0 = tileBase + t * TILE_ROWS;

        stage_tile(data + ((size_t)b * NN + row0) * FF, sData, tid);
        __syncthreads();

        // ---- GEMM #1: x . c^T for this tile (each wave: 16 rows x 32 clusters)
        const float* aRow  = &sData[(wave * 16 + l15) * LDS_STRIDE];
        const float* bRow0 = &sCent[l15 * LDS_STRIDE];
        const float* bRow1 = &sCent[(l15 + 16) * LDS_STRIDE];
        v8f acc0 = {};
        v8f acc1 = {};
        #pragma unroll
        for (int k0 = 0; k0 < FF; k0 += 4) {
            v2f a, b0, b1;
            a.x  = aRow[k0 + koff];   a.y  = aRow[k0 + koff + 1];
            b0.x = bRow0[k0 + koff];  b0.y = bRow0[k0 + koff + 1];
            b1.x = bRow1[k0 + koff];  b1.y = bRow1[k0 + koff + 1];
            acc0 = __builtin_amdgcn_wmma_f32_16x16x4_f32(false, a, false, b0,
                                                         (short)0, acc0, false, false);
            acc1 = __builtin_amdgcn_wmma_f32_16x16x4_f32(false, a, false, b1,
                                                         (short)0, acc1, false, false);
        }

        // Argmin: C layout -> VGPR i = row (i + half*8), lane&15 = cluster column.
        const float c20 = sC2[l15];
        const float c21 = sC2[l15 + 16];
        #pragma unroll
        for (int i = 0; i < 8; ++i) {
            float s  = c20 - 2.0f * acc0[i];
            int   id = l15;
            {
                const float s1 = c21 - 2.0f * acc1[i];
                if (s1 < s) { s = s1; id = l15 + 16; }
            }
            #pragma unroll
            for (int off = 1; off < 16; off <<= 1) {   // xor<16 stays in-half
                const float so = __shfl_xor(s, off, 32);
                const int   io = __shfl_xor(id, off, 32);
                if (so < s || (so == s && io < id)) { s = so; id = io; }
            }
            if (l15 == 0) {
                const int rloc = wave * 16 + i + half * 8;
                out[(size_t)b * NN + tileBase + t * TILE_ROWS + rloc] = id;
                sCid[rloc] = id;
                if (ACCUM) atomicAdd(&sCnt[id], 1.0f);
            }
        }
        __syncthreads();

        // ---- GEMM #2: sums += one_hot(cid)^T (32 x 64) * dataTile (64 x 128)
        if (ACCUM) {
            #pragma unroll
            for (int kc = 0; kc < TILE_ROWS; kc += 4) {
                const int cid0 = sCid[kc + koff];
                const int cid1 = sCid[kc + koff + 1];
                v2f am0, am1;
                am0.x = (cid0 == l15)      ? 1.0f : 0.0f;
                am0.y = (cid1 == l15)      ? 1.0f : 0.0f;
                am1.x = (cid0 == l15 + 16) ? 1.0f : 0.0f;
                am1.y = (cid1 == l15 + 16) ? 1.0f : 0.0f;
                const float* r0 = &sData[(kc + koff)     * LDS_STRIDE];
                const float* r1 = &sData[(kc + koff + 1) * LDS_STRIDE];
                #pragma unroll
                for (int ft = 0; ft < 2; ++ft) {
                    const int f0 = wave * 32 + ft * 16;
                    v2f bv;
                    bv.x = r0[f0 + l15];
                    bv.y = r1[f0 + l15];
                    accS[0][ft] = __builtin_amdgcn_wmma_f32_16x16x4_f32(
                        false, am0, false, bv, (short)0, accS[0][ft], false, false);
                    accS[1][ft] = __builtin_amdgcn_wmma_f32_16x16x4_f32(
                        false, am1, false, bv, (short)0, accS[1][ft], false, false);
                }
            }
        }
        __syncthreads();   // protect sData/sCid before next tile's staging
    }

    // Flush partial sums (exactly K*F atomics per block) and counts.
    if (ACCUM) {
        float* gs = gsums + (size_t)b * KK * FF;
        #pragma unroll
        for (int ct = 0; ct < 2; ++ct) {
            #pragma unroll
            for (int ft = 0; ft < 2; ++ft) {
                const int f = wave * 32 + ft * 16 + l15;
                #pragma unroll
                for (int i = 0; i < 8; ++i) {
                    const int cl = ct * 16 + i + half * 8;
                    atomicAdd(&gs[cl * FF + f], accS[ct][ft][i]);
                }
            }
        }
        if (tid < KK) atomicAdd(&gcounts[b * KK + tid], sCnt[tid]);
    }
}

// ---------------------------------------------------------------------------
__global__ void zero_kernel(float* __restrict__ p, int n)
{
    const int i = blockIdx.x * blockDim.x + threadIdx.x;
    if (i < n) p[i] = 0.0f;
}

// centroids = sums / counts; c2 = ||centroid||^2
__global__ __launch_bounds__(128)
void finalize_kernel(const float* __restrict__ gsums,
                     const float* __restrict__ gcounts,
                     float* __restrict__ cent,
                     float* __restrict__ c2)
{
    const int b = blockIdx.y, k = blockIdx.x, t = threadIdx.x;
    const float cnt = gcounts[b * KK + k];
    const float v = gsums[((size_t)b * KK + k) * FF + t] / cnt;
    cent[((size_t)b * KK + k) * FF + t] = v;

    float p = v * v;
    __shared__ float red[4];
    #pragma unroll
    for (int off = 16; off > 0; off >>= 1) p += __shfl_xor(p, off, 32);
    if ((t & 31) == 0) red[t >> 5] = p;
    __syncthreads();
    if (t == 0) c2[b * KK + k] = red[0] + red[1] + red[2] + red[3];
}

// ---------------------------------------------------------------------------
extern "C" void kernel_launch(void* const* d_in, const int* in_sizes, int n_in,
                              void* d_out, int out_size, void* d_ws, size_t ws_size,
                              hipStream_t stream)
{
    const float* data = (const float*)d_in[0];
    const int*   ids  = (const int*)d_in[1];
    int*         out  = (int*)d_out;

    float* wsf  = (float*)d_ws;
    float* cent = wsf;                          // B*K*F
    float* c2v  = cent + (size_t)BB * KK * FF;  // B*K
    float* sums = c2v + (size_t)BB * KK;        // B*K*F
    float* cnts = sums + (size_t)BB * KK * FF;  // B*K (contiguous after sums)

    init_centroids_kernel<<<dim3(KK, BB), 128, 0, stream>>>(data, ids, cent, c2v);

    const int nz = BB * KK * FF + BB * KK;      // sums + counts (contiguous)
    for (int it = 0; it < ITERS; ++it) {
        zero_kernel<<<(nz + 255) / 256, 256, 0, stream>>>(sums, nz);
        // cid_it = assign(cent_it); sums/counts accumulated with cid_it
        fused_assign_accum_kernel<true><<<dim3(BLOCKS_X, BB), 128, 0, stream>>>(
            data, cent, c2v, out, sums, cnts);
        // cent_{it+1} = sums / counts
        finalize_kernel<<<dim3(KK, BB), 128, 0, stream>>>(sums, cnts, cent, c2v);
    }
    // Final assignment with cent_10 (no accumulation needed).
    fused_assign_accum_kernel<false><<<dim3(BLOCKS_X, BB), 128, 0, stream>>>(
        data, cent, c2v, out, sums, cnts);
}